// TestSMPLLoss_64072322121838
// MI455X (gfx1250) — compile-verified
//
#include <hip/hip_runtime.h>

// MI455X / gfx1250: wave32, WMMA (no MFMA), 320KB LDS per WGP.
typedef __attribute__((ext_vector_type(2))) float v2f;
typedef __attribute__((ext_vector_type(4))) float v4f;
typedef __attribute__((ext_vector_type(8))) float v8f;

#define SDIM  64
#define SS    4096           // SDIM*SDIM
#define BATCH 8
#define PARTS 8              // blocks per batch

// Select d[r] with a cndmask chain (r = lane&7). After this, lanes 0-7 and
// 24-31 hold diagonal elements of the 16x16 WMMA result (C/D layout:
// VGPR r <-> M=r (lanes 0-15) / M=r+8 (lanes 16-31), N = lane%16).
__device__ __forceinline__ float pick8(v8f d, int r) {
  float v = d[0];
  v = (r == 1) ? d[1] : v;
  v = (r == 2) ? d[2] : v;
  v = (r == 3) ? d[3] : v;
  v = (r == 4) ? d[4] : v;
  v = (r == 5) ? d[5] : v;
  v = (r == 6) ? d[6] : v;
  v = (r == 7) ? d[7] : v;
  return v;
}

// Extract diag(D): pixel p's value lives in lane (p<8 ? p : p+16), VGPR p%8.
// Route it to lane p (p = lane&15) with one ds_bpermute. Full-EXEC, no LDS mem.
__device__ __forceinline__ float diag_extract(v8f d, int lane) {
  float v = pick8(d, lane & 7);
  int p = lane & 15;
  int src = (p < 8) ? p : (p + 16);
  return __int_as_float(__builtin_amdgcn_ds_bpermute(src << 2, __float_as_int(v)));
}

__global__ __launch_bounds__(256) void smpl_warp_kernel(
    const float* __restrict__ corr_m,    // [B, 4096, 4096]
    const float* __restrict__ gt_flow,   // [B, 2, 64, 64]
    const float* __restrict__ vis_mask,  // [B, 1, 64, 64]
    const float* __restrict__ scale_ref, // [B, 3, 64, 64]
    float* __restrict__ out)             // [2, B, 3, 64, 64]
{
  __shared__ float sRef[3 * SS];         // 48 KB: ref_v for this batch
  const int tid  = threadIdx.x;
  const int b    = blockIdx.x / PARTS;
  const int part = blockIdx.x % PARTS;

  { // stage ref_v[b] into LDS, 128-bit vectorized
    const v4f* src = (const v4f*)(scale_ref + (size_t)b * 3 * SS);
    v4f* dst = (v4f*)sRef;
#pragma unroll
    for (int i = 0; i < (3 * SS / 4) / 256; ++i)
      dst[tid + i * 256] = src[tid + i * 256];
  }
  __syncthreads();

  const int lane = tid & 31;
  const int wave = tid >> 5;
  const int col  = lane & 15;   // pixel-in-tile (16 pixels per wave per iter)
  const int half = lane >> 4;   // 0: corners {0,1} (K0,K1); 1: corners {2,3} (K2,K3)

  const float* gfx = gt_flow + (size_t)b * 2 * SS;
  const float* gfy = gfx + SS;
  const float* vm  = vis_mask + (size_t)b * SS;
  const size_t corr_base = (size_t)b * SS * SS;
  const v8f zero = {};

  for (int it = 0; it < (SS / PARTS) / 128; ++it) {
    const int n = part * (SS / PARTS) + it * 128 + wave * 16 + col;

    // denormalize flow -> pixel coords (both halves redundantly compute meta)
    const float gx = (gfx[n] + 1.0f) * ((SDIM - 1) * 0.5f);
    const float gy = (gfy[n] + 1.0f) * ((SDIM - 1) * 0.5f);
    const float fx = floorf(gx), fy = floorf(gy);
    const int ix = (int)fx, iy = (int)fy;
    const int x0 = min(max(iy,     0), SDIM - 1);
    const int x1 = min(max(iy + 1, 0), SDIM - 1);
    const int y0 = min(max(ix,     0), SDIM - 1);
    const int y1 = min(max(ix + 1, 0), SDIM - 1);
    const int idx0 = x0 * SDIM + y0, idx1 = x0 * SDIM + y1;
    const int idx2 = x1 * SDIM + y0, idx3 = x1 * SDIM + y1;

    // scatter-with-overwrite semantics: corner i survives only if no later
    // corner lands on the same cell (only possible under clamping)
    const float a0 = (idx0 != idx1 && idx0 != idx2 && idx0 != idx3) ? 1.f : 0.f;
    const float a1 = (idx1 != idx2 && idx1 != idx3) ? 1.f : 0.f;
    const float a2 = (idx2 != idx3) ? 1.f : 0.f;

    const float w0 = (fy + 1.f - gy) * (fx + 1.f - gx) * a0;
    const float w1 = (fy + 1.f - gy) * (gx - fx)       * a1;
    const float w2 = (gy - fy)       * (fx + 1.f - gx) * a2;
    const float w3 = (gy - fy)       * (gx - fx);

    // this lane's corner pair (maps straight onto the WMMA A/B K-split)
    const int   iA = half ? idx2 : idx0;
    const int   iB = half ? idx3 : idx1;
    const float mA = half ? a2 : a0;
    const float mB = half ? 1.f : a1;

    // sparse gather from the 512MB corr tensor: 2 elements of row n per lane
    const float* row = corr_m + corr_base + (size_t)n * SS;
    v2f Aw, Ac;
    Aw.x = half ? w2 : w0;
    Aw.y = half ? w3 : w1;
    Ac.x = row[iA] * mA;
    Ac.y = row[iB] * mB;

    const float vis = vm[n];

#pragma unroll
    for (int c = 0; c < 3; ++c) {
      // B (4x16): column q = pixel q's gathered ref values, gathered from LDS
      v2f Bc;
      Bc.x = sRef[c * SS + iA];
      Bc.y = sRef[c * SS + iB];

      // D = A(16x4) x B(4x16); diag(D)[p] = sum_k val_k(p)*ref[c, idx_k(p)]
      v8f dsm = __builtin_amdgcn_wmma_f32_16x16x4_f32(
          false, Aw, false, Bc, (short)0, zero, false, false);
      v8f dcr = __builtin_amdgcn_wmma_f32_16x16x4_f32(
          false, Ac, false, Bc, (short)0, zero, false, false);

      const float smpl = diag_extract(dsm, lane);
      const float corr = diag_extract(dcr, lane);

      if (lane < 16) { // one writer per pixel
        out[((size_t)(0 * BATCH + b) * 3 + c) * SS + n] = smpl * vis;
        out[((size_t)(1 * BATCH + b) * 3 + c) * SS + n] = corr * vis;
      }
    }
  }
}

extern "C" void kernel_launch(void* const* d_in, const int* in_sizes, int n_in,
                              void* d_out, int out_size, void* d_ws, size_t ws_size,
                              hipStream_t stream) {
  (void)in_sizes; (void)n_in; (void)out_size; (void)d_ws; (void)ws_size;
  const float* corr_m    = (const float*)d_in[0];
  const float* gt_flow   = (const float*)d_in[1];
  const float* vis_mask  = (const float*)d_in[2];
  const float* scale_ref = (const float*)d_in[3];
  float* out = (float*)d_out;

  dim3 grid(BATCH * PARTS), block(256);
  hipLaunchKernelGGL(smpl_warp_kernel, grid, block, 0, stream,
                     corr_m, gt_flow, vis_mask, scale_ref, out);
}